// ExtractLayer_70617852280967
// MI455X (gfx1250) — compile-verified
//
#include <hip/hip_runtime.h>
#include <hip/hip_bf16.h>

// ---------------------------------------------------------------------------
// Multi-scale deformable attention (single level) for MI455X / gfx1250.
// Fixed problem: N=4, Lq=5376, Lin=1024 (32x32), d=768, HEADS=6, HEAD_DIM=128,
// LEVELS=1, POINTS=4.  All GEMM dims divide the 128x128/BK=32 tiling exactly.
// GEMMs use V_WMMA_F32_16X16X4_F32 with async global->LDS tile staging
// (GLOBAL_LOAD_ASYNC_TO_LDS_B128 + s_wait_asynccnt, the CDNA5 ASYNCcnt path).
// ---------------------------------------------------------------------------

typedef float v2f __attribute__((ext_vector_type(2)));
typedef float v8f __attribute__((ext_vector_type(8)));

#define DIM      768
#define HEADS    6
#define HEAD_DIM 128
#define POINTS   4
#define NOFF     48   // HEADS*LEVELS*POINTS*2
#define NATTN    24   // HEADS*LEVELS*POINTS

// ------------------------------- LayerNorm --------------------------------
// One 256-thread block per row of 768 elements (3 per thread).
__global__ void layernorm_kernel(const float* __restrict__ x,
                                 const float* __restrict__ g,
                                 const float* __restrict__ b,
                                 float* __restrict__ y) {
  const int row = blockIdx.x;
  const int tid = threadIdx.x;
  const float* xr = x + (size_t)row * DIM;

  float v[3];
  float s = 0.f, ss = 0.f;
#pragma unroll
  for (int i = 0; i < 3; ++i) {
    v[i] = xr[tid + i * 256];
    s  += v[i];
    ss += v[i] * v[i];
  }

  __shared__ float r1[256];
  __shared__ float r2[256];
  r1[tid] = s;
  r2[tid] = ss;
  __syncthreads();
#pragma unroll
  for (int off = 128; off > 0; off >>= 1) {
    if (tid < off) {
      r1[tid] += r1[tid + off];
      r2[tid] += r2[tid + off];
    }
    __syncthreads();
  }
  const float mu   = r1[0] * (1.0f / DIM);
  const float var  = r2[0] * (1.0f / DIM) - mu * mu;
  const float rstd = rsqrtf(var + 1e-6f);

  float* yr = y + (size_t)row * DIM;
#pragma unroll
  for (int i = 0; i < 3; ++i) {
    const int c = tid + i * 256;
    yr[c] = (v[i] - mu) * rstd * g[c] + b[c];
  }
}

// ----------------------------- WMMA fp32 GEMM -----------------------------
// C[M,Nc] = A[M,K] * W[K,Nc] + bias (+ resid).  M%128==0, K%32==0, Nc%128==0.
// Block: 256 threads = 8 waves; tile 128x128; wave (wm in 0..3, wn in 0..1)
// owns 32 rows x 64 cols = 2x4 accumulators of V_WMMA_F32_16X16X4_F32.
// Tiles are staged global->LDS with GLOBAL_LOAD_ASYNC_TO_LDS_B128 (ASYNCcnt).
template <bool RESID>
__global__ void gemm_wmma(const float* __restrict__ A,
                          const float* __restrict__ W,
                          const float* __restrict__ bias,
                          const float* __restrict__ resid,
                          float* __restrict__ C,
                          int M, int K, int Nc) {
  __shared__ float As[128][36];   // padded: row stride 144B (16B-aligned)
  __shared__ float Bs[32][132];   // padded: row stride 528B (16B-aligned)

  const int tid  = threadIdx.x;
  const int lane = tid & 31;
  const int wave = tid >> 5;
  const int wm   = wave & 3;   // row group (32 rows each)
  const int wn   = wave >> 2;  // col group (64 cols each)

  const int rowBase = blockIdx.y * 128;
  const int colBase = blockIdx.x * 128;

  v8f acc[2][4];
#pragma unroll
  for (int i = 0; i < 2; ++i)
#pragma unroll
    for (int j = 0; j < 4; ++j)
      acc[i][j] = v8f{};

  const int m16  = lane & 15;
  const int khi  = (lane >= 16) ? 2 : 0;   // A/B lane->K split per ISA layout
  const int rhi  = (lane >= 16) ? 8 : 0;   // C/D lane->row split per ISA layout

  for (int k0 = 0; k0 < K; k0 += 32) {
    // ---- stage A tile (128x32): 1024 async B128 ops, 4 per thread ----
#pragma unroll
    for (int i = 0; i < 4; ++i) {
      const int idx = tid + i * 256;
      const int r = idx >> 3;          // 8 x 16B per row
      const int c = (idx & 7) << 2;
      const unsigned ldsA = (unsigned)(uintptr_t)&As[r][c];
      const unsigned gA   = (unsigned)(((size_t)(rowBase + r) * K + (k0 + c))
                                       * sizeof(float));
      asm volatile("global_load_async_to_lds_b128 %0, %1, %2"
                   :: "v"(ldsA), "v"(gA), "s"(A) : "memory");
    }
    // ---- stage B tile (32x128): 1024 async B128 ops, 4 per thread ----
#pragma unroll
    for (int i = 0; i < 4; ++i) {
      const int idx = tid + i * 256;
      const int r = idx >> 5;          // 32 x 16B per row
      const int c = (idx & 31) << 2;
      const unsigned ldsB = (unsigned)(uintptr_t)&Bs[r][c];
      const unsigned gB   = (unsigned)(((size_t)(k0 + r) * Nc + (colBase + c))
                                       * sizeof(float));
      asm volatile("global_load_async_to_lds_b128 %0, %1, %2"
                   :: "v"(ldsB), "v"(gB), "s"(W) : "memory");
    }
    // prefetch next K-chunk of A into L2 (global_prefetch_b8)
    if (k0 + 32 < K)
      __builtin_prefetch(A + (size_t)(rowBase + (tid >> 1)) * K + k0 + 32, 0, 1);

    asm volatile("s_wait_asynccnt 0" ::: "memory");
    __syncthreads();

    // ---- compute: 8 K-steps of 4, 8 WMMAs per step ----
#pragma unroll
    for (int kk = 0; kk < 32; kk += 4) {
      const int kb = kk + khi;
      v2f afr[2];
      const int mrow = wm * 32 + m16;
      afr[0].x = As[mrow][kb];      afr[0].y = As[mrow][kb + 1];
      afr[1].x = As[mrow + 16][kb]; afr[1].y = As[mrow + 16][kb + 1];
      v2f bfr[4];
      const int ncol = wn * 64 + m16;
#pragma unroll
      for (int j = 0; j < 4; ++j) {
        bfr[j].x = Bs[kb][ncol + j * 16];
        bfr[j].y = Bs[kb + 1][ncol + j * 16];
      }
#pragma unroll
      for (int i = 0; i < 2; ++i)
#pragma unroll
        for (int j = 0; j < 4; ++j)
          acc[i][j] = __builtin_amdgcn_wmma_f32_16x16x4_f32(
              false, afr[i], false, bfr[j], (short)0, acc[i][j], false, false);
    }
    __syncthreads();
  }

  // ---- epilogue: bias (+ residual), scatter store ----
#pragma unroll
  for (int i = 0; i < 2; ++i) {
#pragma unroll
    for (int j = 0; j < 4; ++j) {
      const int col = colBase + wn * 64 + j * 16 + m16;
      const float bv = bias[col];
#pragma unroll
      for (int r = 0; r < 8; ++r) {
        const int row = rowBase + wm * 32 + i * 16 + r + rhi;
        float out = acc[i][j][r] + bv;
        if (RESID) out += resid[(size_t)row * Nc + col];
        C[(size_t)row * Nc + col] = out;
      }
    }
  }
}

// ------------------------- offset / attn projections -----------------------
// One block per query row; q row cached in LDS; 72 active threads compute the
// 48 offset + 24 attention-logit dot products (K=768 each).
__global__ void offattn_kernel(const float* __restrict__ q,
                               const float* __restrict__ W_off,
                               const float* __restrict__ b_off,
                               const float* __restrict__ W_attn,
                               const float* __restrict__ b_attn,
                               float* __restrict__ off,
                               float* __restrict__ awl) {
  const int row = blockIdx.x;
  __shared__ float qs[DIM];
  for (int i = threadIdx.x; i < DIM; i += 256)
    qs[i] = q[(size_t)row * DIM + i];
  __syncthreads();

  const int t = threadIdx.x;
  if (t < NOFF) {
    float acc = b_off[t];
#pragma unroll 8
    for (int k = 0; k < DIM; ++k) acc += qs[k] * W_off[(size_t)k * NOFF + t];
    off[(size_t)row * NOFF + t] = acc;
  } else if (t < NOFF + NATTN) {
    const int u = t - NOFF;
    float acc = b_attn[u];
#pragma unroll 8
    for (int k = 0; k < DIM; ++k) acc += qs[k] * W_attn[(size_t)k * NATTN + u];
    awl[(size_t)row * NATTN + u] = acc;
  }
}

// --------------------------- deformable sampling ---------------------------
// One wave per (query-row, head); each lane owns 4 contiguous channels.
// value layout: (N*Lin, 768) with col = head*128 + c (natural GEMM output).
__global__ void msdeform_kernel(const float* __restrict__ value,
                                const float* __restrict__ refp,
                                const float* __restrict__ off,
                                const float* __restrict__ awl,
                                const int* __restrict__ shapes,
                                float* __restrict__ attn_pre,
                                int Lq, int Lin, int totalWaves) {
  const int gw = blockIdx.x * 8 + (threadIdx.x >> 5);
  if (gw >= totalWaves) return;
  const int lane = threadIdx.x & 31;
  const int r = gw / HEADS;
  const int h = gw - r * HEADS;
  const int n = r / Lq;
  const int H_ = shapes[0];
  const int W_ = shapes[1];

  const float rx = refp[(size_t)r * 2 + 0];
  const float ry = refp[(size_t)r * 2 + 1];

  // softmax over the 4 point logits of this head
  float lg[POINTS];
#pragma unroll
  for (int p = 0; p < POINTS; ++p)
    lg[p] = awl[(size_t)r * NATTN + h * POINTS + p];
  float mx = fmaxf(fmaxf(lg[0], lg[1]), fmaxf(lg[2], lg[3]));
  float e[POINTS], sum = 0.f;
#pragma unroll
  for (int p = 0; p < POINTS; ++p) { e[p] = __expf(lg[p] - mx); sum += e[p]; }
  const float inv = 1.0f / sum;

  float ax = 0.f, ay = 0.f, az = 0.f, aw4 = 0.f;
  const size_t chan = (size_t)h * HEAD_DIM + (size_t)lane * 4;

#pragma unroll
  for (int p = 0; p < POINTS; ++p) {
    const float ox = off[(size_t)r * NOFF + h * 8 + p * 2 + 0];
    const float oy = off[(size_t)r * NOFF + h * 8 + p * 2 + 1];
    const float lx = rx + ox / (float)W_;
    const float ly = ry + oy / (float)H_;
    const float px = lx * (float)W_ - 0.5f;
    const float py = ly * (float)H_ - 0.5f;
    const float x0f = floorf(px), y0f = floorf(py);
    const float fx = px - x0f, fy = py - y0f;
    const int x0 = (int)x0f, y0 = (int)y0f;
    const float wgt = e[p] * inv;

#pragma unroll
    for (int dy = 0; dy < 2; ++dy) {
#pragma unroll
      for (int dx = 0; dx < 2; ++dx) {
        const int xi = x0 + dx, yi = y0 + dy;
        if (xi >= 0 && xi < W_ && yi >= 0 && yi < H_) {
          const float w = (dx ? fx : 1.f - fx) * (dy ? fy : 1.f - fy) * wgt;
          const size_t base = ((size_t)(n * Lin + yi * W_ + xi)) * DIM + chan;
          const float4 v = *(const float4*)(value + base);
          ax += w * v.x; ay += w * v.y; az += w * v.z; aw4 += w * v.w;
        }
      }
    }
  }
  float4 o; o.x = ax; o.y = ay; o.z = az; o.w = aw4;
  *(float4*)(attn_pre + (size_t)r * DIM + chan) = o;
}

// -------------------------------- launcher ---------------------------------
extern "C" void kernel_launch(void* const* d_in, const int* in_sizes, int n_in,
                              void* d_out, int out_size, void* d_ws, size_t ws_size,
                              hipStream_t stream) {
  const float* query   = (const float*)d_in[0];
  const float* refp    = (const float*)d_in[1];
  const float* feat    = (const float*)d_in[2];
  const int*   shapes  = (const int*)d_in[3];
  // d_in[4] = level_start_index (unused, single level)
  const float* q_gamma = (const float*)d_in[5];
  const float* q_beta  = (const float*)d_in[6];
  const float* f_gamma = (const float*)d_in[7];
  const float* f_beta  = (const float*)d_in[8];
  const float* W_value = (const float*)d_in[9];
  const float* b_value = (const float*)d_in[10];
  const float* W_off   = (const float*)d_in[11];
  const float* b_off   = (const float*)d_in[12];
  const float* W_attn  = (const float*)d_in[13];
  const float* b_attn  = (const float*)d_in[14];
  const float* W_out   = (const float*)d_in[15];
  const float* b_out   = (const float*)d_in[16];

  const int NLq  = in_sizes[1] / 2;    // N * Lq   (LEVELS==1)
  const int NLin = in_sizes[2] / DIM;  // N * Lin
  const int Nb   = 4;                  // batch size fixed by harness setup
  const int Lq   = NLq / Nb;
  const int Lin  = NLin / Nb;

  // workspace layout (floats)
  float* ws       = (float*)d_ws;
  float* q_ln     = ws;
  float* fln      = q_ln  + (size_t)NLq  * DIM;
  float* value    = fln   + (size_t)NLin * DIM;
  float* offb     = value + (size_t)NLin * DIM;
  float* awlb     = offb  + (size_t)NLq  * NOFF;
  float* attn_pre = awlb  + (size_t)NLq  * NATTN;

  // 1) LayerNorms
  layernorm_kernel<<<NLq,  256, 0, stream>>>(query, q_gamma, q_beta, q_ln);
  layernorm_kernel<<<NLin, 256, 0, stream>>>(feat,  f_gamma, f_beta, fln);

  // 2) value = LN(feat) @ W_value + b_value       (4096x768 * 768x768)
  {
    dim3 grid(DIM / 128, NLin / 128);
    gemm_wmma<false><<<grid, 256, 0, stream>>>(fln, W_value, b_value, nullptr,
                                               value, NLin, DIM, DIM);
  }

  // 3) offset / attention projections
  offattn_kernel<<<NLq, 256, 0, stream>>>(q_ln, W_off, b_off, W_attn, b_attn,
                                          offb, awlb);

  // 4) deformable bilinear sampling + softmax-weighted accumulation
  {
    const int totalWaves = NLq * HEADS;
    msdeform_kernel<<<(totalWaves + 7) / 8, 256, 0, stream>>>(
        value, refp, offb, awlb, shapes, attn_pre, Lq, Lin, totalWaves);
  }

  // 5) out = query + attn_pre @ W_out + b_out     (21504x768 * 768x768)
  {
    dim3 grid(DIM / 128, NLq / 128);
    gemm_wmma<true><<<grid, 256, 0, stream>>>(attn_pre, W_out, b_out, query,
                                              (float*)d_out, NLq, DIM, DIM);
  }
}